// MessagePassing_40054865002690
// MI455X (gfx1250) — compile-verified
//
#include <hip/hip_runtime.h>

typedef __attribute__((ext_vector_type(2))) float v2f;
typedef __attribute__((ext_vector_type(8))) float v8f;

#define BB   4
#define NN   512
#define FN   256
#define FE   8
#define FO   128
#define ROWS (BB * NN)          // 2048 flattened (b, i) rows

// ---------------------------------------------------------------------------
// Kernel 1: copy adj -> out and reduce adj over j -> adjsum[row][e]
// One block per (b,i) row: 512 j * 8 e = 4096 floats = 16KB per row.
// Pure bandwidth kernel (this dominates runtime: ~67MB total traffic).
// ---------------------------------------------------------------------------
__global__ __launch_bounds__(256) void adj_copy_reduce(
    const float* __restrict__ adj, float* __restrict__ adj_out,
    float* __restrict__ adjsum) {
  const int row = blockIdx.x;   // 0..2047
  const int t   = threadIdx.x;  // 0..255
  const float4* __restrict__ src = (const float4*)(adj + (size_t)row * (NN * FE));
  float4* __restrict__ dst       = (float4*)(adj_out + (size_t)row * (NN * FE));

  float acc[8];
#pragma unroll
  for (int e = 0; e < 8; ++e) acc[e] = 0.0f;

#pragma unroll
  for (int rep = 0; rep < 2; ++rep) {
    const int j = t + rep * 256;              // 0..511
    float4 a0 = src[j * 2 + 0];               // e = 0..3
    float4 a1 = src[j * 2 + 1];               // e = 4..7
    dst[j * 2 + 0] = a0;
    dst[j * 2 + 1] = a1;
    acc[0] += a0.x; acc[1] += a0.y; acc[2] += a0.z; acc[3] += a0.w;
    acc[4] += a1.x; acc[5] += a1.y; acc[6] += a1.z; acc[7] += a1.w;
  }

  __shared__ float red[256][9];               // pad to 9 to dodge bank conflicts
#pragma unroll
  for (int e = 0; e < 8; ++e) red[t][e] = acc[e];
  __syncthreads();

#pragma unroll
  for (int s = 128; s > 0; s >>= 1) {
    if (t < s) {
#pragma unroll
      for (int e = 0; e < 8; ++e) red[t][e] += red[t + s][e];
    }
    __syncthreads();
  }
  if (t < 8) adjsum[row * FE + t] = red[0][t];
}

// ---------------------------------------------------------------------------
// Kernel 2: h = x @ W_node  (2048x256 @ 256x128), f32 WMMA 16x16x4.
// One 16x16 output tile per wave; K-loop of 64 WMMAs.
// Fragment layouts per ISA 7.12.2:
//   A (16x4): lane l<16 holds row l, {K, K+1}; lane 16+l holds row l, {K+2, K+3}
//   B (4x16): lane l<16 holds col l, {K, K+1}; lane 16+l holds col l, {K+2, K+3}
//   C/D     : vgpr v -> row v (lanes 0-15) / row v+8 (lanes 16-31), col = l
// ---------------------------------------------------------------------------
__global__ __launch_bounds__(256) void gemm_node(
    const float* __restrict__ X, const float* __restrict__ Wn,
    float* __restrict__ H) {
  const int wave = threadIdx.x >> 5;          // 0..7
  const int lane = threadIdx.x & 31;
  const int tile = blockIdx.x * 8 + wave;     // 0..1023
  const int col0 = (tile & 7) * 16;           // 128/16 = 8 col tiles
  const int row0 = (tile >> 3) * 16;          // 2048/16 = 128 row tiles
  const int half = lane >> 4;
  const int l    = lane & 15;

  v8f c = {};
  for (int k = 0; k < FN; k += 4) {
    const int kb = k + 2 * half;
    v2f a, b;
    a.x = X[(size_t)(row0 + l) * FN + kb];
    a.y = X[(size_t)(row0 + l) * FN + kb + 1];
    b.x = Wn[(size_t)kb * FO + col0 + l];
    b.y = Wn[(size_t)(kb + 1) * FO + col0 + l];
    c = __builtin_amdgcn_wmma_f32_16x16x4_f32(false, a, false, b,
                                              (short)0, c, false, false);
  }
#pragma unroll
  for (int v = 0; v < 8; ++v)
    H[(size_t)(row0 + v + 8 * half) * FO + col0 + l] = c[v];
}

// ---------------------------------------------------------------------------
// Kernel 3: msg_j[b][p] = (sum_i h[b,i,:]) @ W_j   (W_j = W_edge rows 128..255)
// ---------------------------------------------------------------------------
__global__ __launch_bounds__(128) void msgj_kernel(
    const float* __restrict__ H, const float* __restrict__ Wedge,
    float* __restrict__ msgj) {
  const int b = blockIdx.x;    // 0..3
  const int o = threadIdx.x;   // 0..127
  const float* __restrict__ hb = H + (size_t)b * NN * FO;
  float s = 0.0f;
  for (int i = 0; i < NN; ++i) s += hb[(size_t)i * FO + o];
  __shared__ float hs[FO];
  hs[o] = s;
  __syncthreads();
  float m = 0.0f;
  for (int oo = 0; oo < FO; ++oo) m += hs[oo] * Wedge[(size_t)(FO + oo) * FO + o];
  msgj[b * FO + o] = m;
}

// ---------------------------------------------------------------------------
// Kernel 4: new_x = 512*(h @ W_i) + msg_j + adjsum @ W_e + 512*b_edge + h
// WMMA for h @ W_i (K=128), fused scalar epilogue on the accumulator frags.
// ---------------------------------------------------------------------------
__global__ __launch_bounds__(256) void final_kernel(
    const float* __restrict__ H, const float* __restrict__ Wedge,
    const float* __restrict__ bedge, const float* __restrict__ adjsum,
    const float* __restrict__ msgj, float* __restrict__ out_x) {
  const int wave = threadIdx.x >> 5;
  const int lane = threadIdx.x & 31;
  const int tile = blockIdx.x * 8 + wave;     // 0..1023
  const int col0 = (tile & 7) * 16;
  const int row0 = (tile >> 3) * 16;
  const int half = lane >> 4;
  const int l    = lane & 15;

  v8f c = {};
  for (int k = 0; k < FO; k += 4) {           // W_i = W_edge rows 0..127
    const int kb = k + 2 * half;
    v2f a, b;
    a.x = H[(size_t)(row0 + l) * FO + kb];
    a.y = H[(size_t)(row0 + l) * FO + kb + 1];
    b.x = Wedge[(size_t)kb * FO + col0 + l];
    b.y = Wedge[(size_t)(kb + 1) * FO + col0 + l];
    c = __builtin_amdgcn_wmma_f32_16x16x4_f32(false, a, false, b,
                                              (short)0, c, false, false);
  }

  const int p = col0 + l;
  const float bias = 512.0f * bedge[p];
#pragma unroll
  for (int v = 0; v < 8; ++v) {
    const int row = row0 + v + 8 * half;
    const int b   = row >> 9;                 // 512 rows per batch
    float me = 0.0f;
#pragma unroll
    for (int e = 0; e < 8; ++e)
      me += adjsum[row * FE + e] * Wedge[(size_t)(2 * FO + e) * FO + p];
    out_x[(size_t)row * FO + p] =
        512.0f * c[v] + msgj[b * FO + p] + me + bias + H[(size_t)row * FO + p];
  }
}

// ---------------------------------------------------------------------------
extern "C" void kernel_launch(void* const* d_in, const int* in_sizes, int n_in,
                              void* d_out, int out_size, void* d_ws,
                              size_t ws_size, hipStream_t stream) {
  const float* x     = (const float*)d_in[0];  // (4,512,256)
  const float* adj   = (const float*)d_in[1];  // (4,512,512,8)
  const float* Wn    = (const float*)d_in[2];  // (256,128)
  const float* Wedge = (const float*)d_in[3];  // (264,128)
  const float* be    = (const float*)d_in[4];  // (128,)

  float* out_x   = (float*)d_out;              // 2048*128 floats
  float* out_adj = out_x + (size_t)ROWS * FO;  // adj copy, 8.39M floats

  float* H      = (float*)d_ws;                        // 2048*128
  float* adjsum = H + (size_t)ROWS * FO;               // 2048*8
  float* msgj   = adjsum + (size_t)ROWS * FE;          // 4*128

  // 1) adj copy + j-reduction (bandwidth dominated: ~67MB)
  adj_copy_reduce<<<ROWS, 256, 0, stream>>>(adj, out_adj, adjsum);
  // 2) h = x @ W_node (WMMA f32)
  gemm_node<<<(ROWS / 16) * (FO / 16) / 8, 256, 0, stream>>>(x, Wn, H);
  // 3) msg_j per batch
  msgj_kernel<<<BB, FO, 0, stream>>>(H, Wedge, msgj);
  // 4) fused output (WMMA f32 + epilogue)
  final_kernel<<<(ROWS / 16) * (FO / 16) / 8, 256, 0, stream>>>(
      H, Wedge, be, adjsum, msgj, out_x);
}